// GActor_29137058136595
// MI455X (gfx1250) — compile-verified
//
#include <hip/hip_runtime.h>

typedef float v2f __attribute__((ext_vector_type(2)));
typedef float v8f __attribute__((ext_vector_type(8)));

#define STATE 64
#define HID 64

// ---------------- degree / normalization ----------------
__global__ void k_init_deg(float* __restrict__ deg, int n) {
    int i = blockIdx.x * blockDim.x + threadIdx.x;
    if (i < n) deg[i] = 1.0f;                     // self-loop contributes 1
}

__global__ void k_deg_edges(const long long* __restrict__ col, float* __restrict__ deg, int e) {
    int i = blockIdx.x * blockDim.x + threadIdx.x;
    if (i < e) atomicAdd(&deg[(int)col[i]], 1.0f);
}

__global__ void k_rsqrt(float* __restrict__ d, int n) {
    int i = blockIdx.x * blockDim.x + threadIdx.x;
    if (i < n) d[i] = rsqrtf(d[i]);               // deg >= 1 always
}

// ---------------- utility ----------------
__global__ void k_zero(float* __restrict__ p, long long n) {
    long long i = (long long)blockIdx.x * blockDim.x + threadIdx.x;
    if (i < n) p[i] = 0.0f;
}

// ---------------- H = X @ W  (N x 64 x 64), f32 WMMA ----------------
// Wave w of the 128-thread block computes the 16x16 tile at cols [16w,16w+16),
// rows [16*blockIdx.x, +16). W (64x64) staged in LDS.
__global__ void __launch_bounds__(128) k_gemm64(const float* __restrict__ X,
                                                const float* __restrict__ W,
                                                float* __restrict__ H, int n) {
    __shared__ float Wl[STATE * HID];
    int t = threadIdx.x;
#pragma unroll
    for (int i = 0; i < (STATE * HID) / 128; ++i)
        Wl[t + 128 * i] = W[t + 128 * i];
    __syncthreads();

    int wave = t >> 5;
    int lane = t & 31;
    int half = lane >> 4;        // 0: K pair {k0,k0+1}; 1: K pair {k0+2,k0+3}
    int lm   = lane & 15;        // row within A tile / col within B,D tile
    int rowBase = blockIdx.x * 16;
    int colBase = wave * 16;

    int arow = rowBase + lm;
    if (arow >= n) arow = n - 1;              // clamp loads, keep EXEC all-ones
    const float* xrow = X + (size_t)arow * STATE;

    v8f acc = {};
#pragma unroll
    for (int k0 = 0; k0 < STATE; k0 += 4) {
        int ka = k0 + 2 * half;
        // A fragment: A[M=lm][ka], A[M=lm][ka+1]  (8B-aligned pair)
        v2f a = *(const v2f*)(xrow + ka);
        // B fragment: B[K=ka][N=colBase+lm], B[K=ka+1][N=colBase+lm] (LDS)
        v2f b;
        b.x = Wl[ka * HID + colBase + lm];
        b.y = Wl[(ka + 1) * HID + colBase + lm];
        acc = __builtin_amdgcn_wmma_f32_16x16x4_f32(false, a, false, b,
                                                    (short)0, acc, false, false);
    }

    // D layout: element r of acc -> row (rowBase + r + 8*half), col (colBase+lm)
#pragma unroll
    for (int r = 0; r < 8; ++r) {
        int orow = rowBase + r + 8 * half;
        if (orow < n)
            H[(size_t)orow * HID + colBase + lm] = acc[r];
    }
}

// ---------------- edge scatter, 64-wide: OUT[col] += H[row] * dis[row]*dis[col] ----------------
// One wave per edge: lane f handles features {2f, 2f+1}; coalesced 256B gather,
// per-element global_atomic_add_f32 into L2-resident OUT.
__global__ void k_edge64(const float* __restrict__ H,
                         const long long* __restrict__ row,
                         const long long* __restrict__ col,
                         const float* __restrict__ dis,
                         float* __restrict__ OUT, int e) {
    long long idx = (long long)blockIdx.x * blockDim.x + threadIdx.x;
    int ei = (int)(idx >> 5);
    int lane = (int)(idx & 31);
    if (ei >= e) return;
    int r = (int)row[ei];
    int c = (int)col[ei];
    float nrm = dis[r] * dis[c];
    v2f v = *(const v2f*)(H + (size_t)r * HID + 2 * lane);
    float* o = OUT + (size_t)c * HID + 2 * lane;
    atomicAdd(o,     v.x * nrm);
    atomicAdd(o + 1, v.y * nrm);
}

// ---------------- finalize: add self-loop + bias (+ReLU), in place ----------------
__global__ void k_fin64(float* __restrict__ AGG, const float* __restrict__ H,
                        const float* __restrict__ dis, const float* __restrict__ bias,
                        int n, int do_relu) {
    long long idx = (long long)blockIdx.x * blockDim.x + threadIdx.x;
    if (idx >= (long long)n * HID) return;
    int v = (int)(idx >> 6);
    int f = (int)(idx & 63);
    float d = dis[v];
    float val = AGG[idx] + H[idx] * d * d + bias[f];
    AGG[idx] = do_relu ? fmaxf(val, 0.0f) : val;
}

// ---------------- y = X @ w  (64 -> 1), wave-per-node ----------------
__global__ void k_gemv(const float* __restrict__ X, const float* __restrict__ w,
                       float* __restrict__ y, int n) {
    long long t = (long long)blockIdx.x * blockDim.x + threadIdx.x;
    int v = (int)(t >> 5);
    int lane = (int)(t & 31);
    if (v >= n) return;
    v2f xv = *(const v2f*)(X + (size_t)v * HID + 2 * lane);
    v2f wv = *(const v2f*)(w + 2 * lane);
    float p = xv.x * wv.x + xv.y * wv.y;
#pragma unroll
    for (int off = 16; off > 0; off >>= 1)
        p += __shfl_xor(p, off, 32);
    if (lane == 0) y[v] = p;
}

// ---------------- edge scatter, 1-wide ----------------
__global__ void k_edge1(const float* __restrict__ h3,
                        const long long* __restrict__ row,
                        const long long* __restrict__ col,
                        const float* __restrict__ dis,
                        float* __restrict__ out, int e) {
    int i = blockIdx.x * blockDim.x + threadIdx.x;
    if (i >= e) return;
    int r = (int)row[i];
    int c = (int)col[i];
    atomicAdd(&out[c], h3[r] * dis[r] * dis[c]);
}

__global__ void k_fin1(float* __restrict__ out, const float* __restrict__ h3,
                       const float* __restrict__ dis, const float* __restrict__ b3, int n) {
    int v = blockIdx.x * blockDim.x + threadIdx.x;
    if (v >= n) return;
    float d = dis[v];
    out[v] += h3[v] * d * d + b3[0];
}

extern "C" void kernel_launch(void* const* d_in, const int* in_sizes, int n_in,
                              void* d_out, int out_size, void* d_ws, size_t ws_size,
                              hipStream_t stream) {
    const float*     x   = (const float*)d_in[0];
    const long long* ei  = (const long long*)d_in[1];
    const float*     W1  = (const float*)d_in[2];
    const float*     b1  = (const float*)d_in[3];
    const float*     W2  = (const float*)d_in[4];
    const float*     b2  = (const float*)d_in[5];
    const float*     W3  = (const float*)d_in[6];
    const float*     b3  = (const float*)d_in[7];
    float*           out = (float*)d_out;

    int n = in_sizes[0] / STATE;   // 100000
    int e = in_sizes[1] / 2;       // 1600000
    const long long* row = ei;         // edge_index[0] = source
    const long long* col = ei + e;     // edge_index[1] = target

    // workspace carve-up (floats): dis[n] | bufA[n*64] | bufB[n*64] | h3[n]
    float* dis  = (float*)d_ws;
    float* bufA = dis + n;
    float* bufB = bufA + (size_t)n * HID;
    float* h3   = bufB + (size_t)n * HID;

    const int B = 256;
    long long nh = (long long)n * HID;
    unsigned gN   = (unsigned)((n + B - 1) / B);
    unsigned gE   = (unsigned)((e + B - 1) / B);
    unsigned gNH  = (unsigned)((nh + B - 1) / B);
    unsigned gE32 = (unsigned)(((long long)e * 32 + B - 1) / B);
    unsigned gN32 = (unsigned)(((long long)n * 32 + B - 1) / B);
    unsigned gT   = (unsigned)((n + 15) / 16);

    // normalization: deg = 1 + in-degree; dis = rsqrt(deg)
    k_init_deg<<<gN, B, 0, stream>>>(dis, n);
    k_deg_edges<<<gE, B, 0, stream>>>(col, dis, e);
    k_rsqrt<<<gN, B, 0, stream>>>(dis, n);

    // layer 1: h1 = x@W1 ; agg ; relu(agg + self + b1)
    k_gemm64<<<gT, 128, 0, stream>>>(x, W1, bufA, n);
    k_zero<<<gNH, B, 0, stream>>>(bufB, nh);
    k_edge64<<<gE32, B, 0, stream>>>(bufA, row, col, dis, bufB, e);
    k_fin64<<<gNH, B, 0, stream>>>(bufB, bufA, dis, b1, n, 1);

    // layer 2: h2 = x2@W2 ; agg ; relu(agg + self + b2)
    k_gemm64<<<gT, 128, 0, stream>>>(bufB, W2, bufA, n);
    k_zero<<<gNH, B, 0, stream>>>(bufB, nh);
    k_edge64<<<gE32, B, 0, stream>>>(bufA, row, col, dis, bufB, e);
    k_fin64<<<gNH, B, 0, stream>>>(bufB, bufA, dis, b2, n, 1);

    // layer 3: h3 = x3@W3 (64->1) ; agg into out ; + self + b3
    k_gemv<<<gN32, B, 0, stream>>>(bufB, W3, h3, n);
    k_zero<<<gN, B, 0, stream>>>(out, n);
    k_edge1<<<gE, B, 0, stream>>>(h3, row, col, dis, out, e);
    k_fin1<<<gN, B, 0, stream>>>(out, h3, dis, b3, n);
}